// ObservedSimpleNMS_15960098472440
// MI455X (gfx1250) — compile-verified
//
#include <hip/hip_runtime.h>
#include <stdint.h>

// SimpleNMS (radius 4, 2 iterations) on 16 x 1024 x 1024 f32.
// Bandwidth-bound stencil (no GEMM -> WMMA inapplicable). CDNA5 paths:
// GLOBAL_LOAD_ASYNC_TO_LDS_B128 (ASYNCcnt) tile loads + s_wait_asynccnt,
// LDS separable 9x9 pooling. Fused per-iteration kernel keeps the
// supp_scores field entirely in LDS: ~0.39 GB total HBM traffic.

#define R     4
#define TS    64
#define HALO  (TS + 2 * R)       // 72  (score/ss tile with pool halo)
#define CH    (HALO / 4)         // 18  16-byte chunks per score row
#define MH    (TS + 4 * R)       // 80  (mask tile: dilate + pool halo)
#define MCH   (MH / 4)           // 20  4-byte chunks per mask row
#define HH    1024
#define WW    1024
#define NEG_INF (-__builtin_huge_valf())

__device__ __forceinline__ uint32_t lds_off_of(const void* p) {
  // generic (flat) pointer to LDS: low 32 bits == LDS byte offset
  return (uint32_t)(uintptr_t)p;
}

__device__ __forceinline__ void async_load_b128(uint32_t lds_off, const void* gptr) {
  asm volatile("global_load_async_to_lds_b128 %0, %1, off"
               :: "v"(lds_off), "v"((uint64_t)(uintptr_t)gptr)
               : "memory");
}

__device__ __forceinline__ void wait_async0() {
  asm volatile("s_wait_asynccnt 0" ::: "memory");
}

// K1: mask[g] = (scores[g] == maxpool9(scores)[g])
__global__ __launch_bounds__(256)
void nms_pool_kernel(const float* __restrict__ in,
                     unsigned char* __restrict__ mask)
{
  __shared__ __align__(16) float tile[HALO][HALO];  // raw scores, -inf OOB
  __shared__ __align__(16) float rowm[HALO][TS];    // clamped horizontal 9-max

  const int tid = threadIdx.x;
  const int tx0 = blockIdx.x * TS;
  const int ty0 = blockIdx.y * TS;
  const size_t ibase = (size_t)blockIdx.z * (size_t)(HH * WW);
  const float* img = in + ibase;

  for (int i = tid; i < HALO * CH; i += 256) {
    const int ly = i / CH;
    const int c  = i - ly * CH;
    const int gy = ty0 - R + ly;
    const int gx = tx0 - R + c * 4;
    float* lp = &tile[ly][c * 4];
    const bool inb = (gy >= 0) & (gy < HH) & (gx >= 0) & (gx + 3 < WW);
    if (inb) {
      async_load_b128(lds_off_of(lp), img + (size_t)gy * WW + gx);   // ASYNCcnt
    } else {
      lp[0] = NEG_INF; lp[1] = NEG_INF; lp[2] = NEG_INF; lp[3] = NEG_INF;
    }
  }
  wait_async0();
  __syncthreads();

  for (int i = tid; i < HALO * TS; i += 256) {
    const int ly = i / TS;
    const int ox = i - ly * TS;
    float m = 0.0f;                     // clamp at 0: exact (values >= 0)
#pragma unroll
    for (int k = 0; k < 9; ++k) m = fmaxf(m, tile[ly][ox + k]);
    rowm[ly][ox] = m;
  }
  __syncthreads();

  unsigned char* mimg = mask + ibase;
  for (int i = tid; i < TS * TS; i += 256) {
    const int oy = i / TS;
    const int ox = i - oy * TS;
    float m = rowm[oy][ox];
#pragma unroll
    for (int k = 1; k < 9; ++k) m = fmaxf(m, rowm[oy + k][ox]);
    const float raw = tile[oy + R][ox + R];
    mimg[(size_t)(ty0 + oy) * WW + (tx0 + ox)] =
        (raw == m) ? (unsigned char)1 : (unsigned char)0;
  }
}

// Fused iteration: supp = dilate9(mask); ss = supp ? -1 : scores (in LDS only);
// new = (!supp && ss == maxpool9(max(ss,0))).
//   write_out == 0: mask |= new
//   write_out == 1: out = (old_mask | new) ? scores : 0   (final gather fused)
__global__ __launch_bounds__(256)
void nms_iter_kernel(unsigned char* __restrict__ mask,
                     const float* __restrict__ scores,
                     float* __restrict__ out,
                     int write_out)
{
  __shared__ __align__(4)  unsigned char mt[MH][MH];    // 80x80 old mask, 0 OOB
  __shared__ __align__(4)  unsigned char ro[MH][HALO];  // 80x72 horizontal OR
  __shared__ __align__(16) float sst[HALO][HALO];       // 72x72 scores -> ss
  __shared__ __align__(16) float rowm[HALO][TS];        // 72x64 horizontal max

  const int tid = threadIdx.x;
  const int tx0 = blockIdx.x * TS;
  const int ty0 = blockIdx.y * TS;
  const size_t ibase = (size_t)blockIdx.z * (size_t)(HH * WW);
  const float* simg = scores + ibase;
  unsigned char* mimg = mask + ibase;

  // A1: async scores -> sst (raw; -inf OOB). 16B chunks fully in/out of bounds.
  for (int i = tid; i < HALO * CH; i += 256) {
    const int ly = i / CH;
    const int c  = i - ly * CH;
    const int gy = ty0 - R + ly;
    const int gx = tx0 - R + c * 4;
    float* lp = &sst[ly][c * 4];
    const bool inb = (gy >= 0) & (gy < HH) & (gx >= 0) & (gx + 3 < WW);
    if (inb) {
      async_load_b128(lds_off_of(lp), simg + (size_t)gy * WW + gx);  // ASYNCcnt
    } else {
      lp[0] = NEG_INF; lp[1] = NEG_INF; lp[2] = NEG_INF; lp[3] = NEG_INF;
    }
  }
  // A2: old mask 80x80 (byte field, 4B chunks fully in/out of bounds)
  for (int i = tid; i < MH * MCH; i += 256) {
    const int ly = i / MCH;
    const int c  = i - ly * MCH;
    const int gy = ty0 - 2 * R + ly;
    const int gx = tx0 - 2 * R + c * 4;
    const bool inb = (gy >= 0) & (gy < HH) & (gx >= 0) & (gx + 3 < WW);
    uint32_t w = 0u;
    if (inb) w = *(const uint32_t*)(mimg + (size_t)gy * WW + gx);
    *(uint32_t*)&mt[ly][c * 4] = w;
  }
  wait_async0();
  __syncthreads();

  // B: horizontal 9-wide OR of mask
  for (int i = tid; i < MH * HALO; i += 256) {
    const int ly = i / HALO;
    const int x  = i - ly * HALO;
    unsigned m = 0;
#pragma unroll
    for (int k = 0; k < 9; ++k) m |= mt[ly][x + k];
    ro[ly][x] = (unsigned char)m;
  }
  __syncthreads();

  // C: vertical OR -> suppression; encode suppressed as -1.0 in sst (exact:
  // max(sst,0) reproduces supp_scores bit-for-bit since scores >= 0)
  for (int i = tid; i < HALO * HALO; i += 256) {
    const int ly = i / HALO;
    const int x  = i - ly * HALO;
    unsigned m = 0;
#pragma unroll
    for (int k = 0; k < 9; ++k) m |= ro[ly + k][x];
    if (m) sst[ly][x] = -1.0f;
  }
  __syncthreads();

  // D: horizontal 9-max of ss, clamped at 0
  for (int i = tid; i < HALO * TS; i += 256) {
    const int ly = i / TS;
    const int ox = i - ly * TS;
    float m = 0.0f;
#pragma unroll
    for (int k = 0; k < 9; ++k) m = fmaxf(m, sst[ly][ox + k]);
    rowm[ly][ox] = m;
  }
  __syncthreads();

  // E: vertical 9-max + epilogue
  for (int i = tid; i < TS * TS; i += 256) {
    const int oy = i / TS;
    const int ox = i - oy * TS;
    float m = rowm[oy][ox];
#pragma unroll
    for (int k = 1; k < 9; ++k) m = fmaxf(m, rowm[oy + k][ox]);
    const float raw = sst[oy + R][ox + R];          // -1 iff suppressed
    const bool nm = (!(raw < 0.0f)) && (raw == m);  // new & ~supp
    const size_t g = ibase + (size_t)(ty0 + oy) * WW + (tx0 + ox);
    if (write_out) {
      const bool fin = nm || (mt[oy + 2 * R][ox + 2 * R] != 0);  // old | new
      out[g] = fin ? scores[g] : 0.0f;
    } else {
      if (nm) mask[g] = 1;                          // set-only update
    }
  }
}

extern "C" void kernel_launch(void* const* d_in, const int* in_sizes, int n_in,
                              void* d_out, int out_size, void* d_ws, size_t ws_size,
                              hipStream_t stream)
{
  const float* scores = (const float*)d_in[0];
  float* out = (float*)d_out;
  unsigned char* mask = (unsigned char*)d_ws;       // B*H*W bytes = 16 MiB

  const int n = in_sizes[0];                        // 16 * 1024 * 1024
  const int Bimg = n / (HH * WW);

  dim3 grid(WW / TS, HH / TS, Bimg);
  dim3 block(256);

  // K1: initial local-max mask (fully overwrites mask -> deterministic)
  nms_pool_kernel<<<grid, block, 0, stream>>>(scores, mask);
  // Iteration 1: update mask in place
  nms_iter_kernel<<<grid, block, 0, stream>>>(mask, scores, out, 0);
  // Iteration 2: fused final gather into d_out
  nms_iter_kernel<<<grid, block, 0, stream>>>(mask, scores, out, 1);
}